// FLG_9088150798943
// MI455X (gfx1250) — compile-verified
//
#include <hip/hip_runtime.h>
#include <hip/hip_bf16.h>
#include <math.h>

// ---------------------------------------------------------------------------
// FLG forward for MI455X (gfx1250, wave32).
//   Stage A: W[e] = prod_w (I - u_w u_w^T), u = sqrt(2)*gelu(v)/||gelu(v)||
//            computed via rank-1 updates (32x fewer FLOPs than dense chain).
//   Stage B: out[b,:,1:] = x[b,:,1:] @ W[r_idx[b]] using V_WMMA_F32_16X16X4_F32
//            (f32 matrix pipe -> matches reference precision; kernel is
//            HBM-bound at ~0.3 GB total traffic anyway).
// ---------------------------------------------------------------------------

#define NUM_EMB 512
#define DM      64          // d = DIM-1
#define DIMF    65
#define BB      2048
#define NN      256

typedef __attribute__((ext_vector_type(2))) float v2f;
typedef __attribute__((ext_vector_type(8))) float v8f;

// ---------------------------------------------------------------------------
// Stage A: one 64-thread block per embedding. Thread t owns row t of M (in
// VGPRs). u rows staged in LDS (16 KB). 64 rank-1 steps:
//   p = <row, u_w>;  row -= p * u_w
// ---------------------------------------------------------------------------
__global__ __launch_bounds__(64) void flg_build_w(
    const float* __restrict__ emb,      // (512, 4096)
    const float* __restrict__ flip,     // scalar
    float* __restrict__ W)              // (512, 64, 64) scratch
{
    __shared__ float u[DM][DM];
    const int e = blockIdx.x;
    const int t = threadIdx.x;          // row owned by this thread

    // gelu (exact, erf-based) + normalize row t -> u[t][:]
    const float* er = emb + (size_t)e * (DM * DM) + (size_t)t * DM;
    float g[DM];
    float sq = 0.f;
#pragma unroll
    for (int i = 0; i < DM; ++i) {
        float xv = er[i];
        float gv = 0.5f * xv * (1.0f + erff(xv * 0.70710678118654752440f));
        g[i] = gv;
        sq += gv * gv;
    }
    const float s = sqrtf(2.0f / sq);
#pragma unroll
    for (int i = 0; i < DM; ++i) u[t][i] = g[i] * s;
    __syncthreads();

    // M = I, then chain the 64 reflections (right-multiplication)
    float m[DM];
#pragma unroll
    for (int c = 0; c < DM; ++c) m[c] = (c == t) ? 1.0f : 0.0f;

    for (int w = 0; w < DM; ++w) {
        float p = 0.f;
#pragma unroll
        for (int c = 0; c < DM; ++c) p += m[c] * u[w][c];   // LDS broadcast
#pragma unroll
        for (int c = 0; c < DM; ++c) m[c] -= p * u[w][c];
    }

    m[DM - 1] *= flip[0];               // W[:,:,-1] *= flip_sign

    float* wr = W + (size_t)e * (DM * DM) + (size_t)t * DM;
#pragma unroll
    for (int c = 0; c < DM; ++c) wr[c] = m[c];
}

// ---------------------------------------------------------------------------
// Stage B: one 256-thread block (8 waves) per batch b.
//   sW (16 KB LDS) <- W[r_idx[b]]
//   wave w computes rows [32w, 32w+32): 2 row-tiles x 4 col-tiles of 16x16,
//   each via 16 chained v_wmma_f32_16x16x4_f32 (K=64).
// WMMA f32 16x16x4 lane mapping (wave32):
//   A: lane holds rows m=lane&15, k-pair {kb+2h, kb+2h+1}, h=lane>>4
//   B: same k-pair, column n=lane&15
//   C/D: vgpr r -> (row r + 8h, col lane&15)
// ---------------------------------------------------------------------------
__global__ __launch_bounds__(256) void flg_apply(
    const float* __restrict__ x,        // (2048, 256, 65)
    const int*   __restrict__ r_idx,    // (2048,)
    const float* __restrict__ W,        // (512, 64, 64)
    float*       __restrict__ out)      // (2048*256*65) xo, then 2048 r_idx
{
    __shared__ float sW[DM * DM];

    const int b   = blockIdx.x;
    const int tid = threadIdx.x;
    const int ridx = r_idx[b];

    const float* Wb = W + (size_t)ridx * (DM * DM);
#pragma unroll 4
    for (int i = tid; i < DM * DM; i += 256) sW[i] = Wb[i];

    const float* xb = x   + (size_t)b * NN * DIMF;
    float*       ob = out + (size_t)b * NN * DIMF;

    // passthrough channel 0
    if (tid < NN) ob[(size_t)tid * DIMF] = xb[(size_t)tid * DIMF];
    // tuple tail: r_idx as numeric values after xo
    if (tid == 0) out[(size_t)BB * NN * DIMF + b] = (float)ridx;

    __syncthreads();

    const int wave = tid >> 5;          // 0..7
    const int lane = tid & 31;
    const int l15  = lane & 15;
    const int h    = lane >> 4;         // 0 or 1

    for (int rt = 0; rt < 2; ++rt) {
        const int m0 = wave * 32 + rt * 16;

        // A tile (16 rows x K=64) into registers, reused over 4 col tiles
        v2f areg[16];
        const float* arow = xb + (size_t)(m0 + l15) * DIMF + 1;  // skip x0
#pragma unroll
        for (int ks = 0; ks < 16; ++ks) {
            const int k = ks * 4 + 2 * h;
            areg[ks].x = arow[k];
            areg[ks].y = arow[k + 1];
        }

#pragma unroll
        for (int ct = 0; ct < 4; ++ct) {
            const int n0 = ct * 16;
            v8f acc = {};
#pragma unroll
            for (int ks = 0; ks < 16; ++ks) {
                const int k = ks * 4 + 2 * h;
                v2f bv;
                bv.x = sW[(k)     * DM + n0 + l15];
                bv.y = sW[(k + 1) * DM + n0 + l15];
                acc = __builtin_amdgcn_wmma_f32_16x16x4_f32(
                    /*neg_a=*/false, areg[ks],
                    /*neg_b=*/false, bv,
                    /*c_mod=*/(short)0, acc,
                    /*reuse_a=*/false, /*reuse_b=*/false);
            }
            // scatter 16x16 D tile: vgpr r -> row m0+r+8h, col n0+l15, chan+1
#pragma unroll
            for (int r = 0; r < 8; ++r) {
                ob[(size_t)(m0 + r + 8 * h) * DIMF + 1 + n0 + l15] = acc[r];
            }
        }
    }
}

extern "C" void kernel_launch(void* const* d_in, const int* in_sizes, int n_in,
                              void* d_out, int out_size, void* d_ws, size_t ws_size,
                              hipStream_t stream) {
    const float* x     = (const float*)d_in[0];   // (2048,256,65) f32
    const int*   r_idx = (const int*)  d_in[1];   // (2048,) i32
    const float* emb   = (const float*)d_in[2];   // (512,4096) f32
    const float* flip  = (const float*)d_in[3];   // scalar f32

    float* W = (float*)d_ws;                      // 512*64*64*4 = 8 MB scratch

    flg_build_w<<<NUM_EMB, 64, 0, stream>>>(emb, flip, W);
    flg_apply  <<<BB, 256, 0, stream>>>(x, r_idx, W, (float*)d_out);
}